// RelationalGATLayer_36661840839323
// MI455X (gfx1250) — compile-verified
//
#include <hip/hip_runtime.h>
#include <hip/hip_bf16.h>

typedef float v2f __attribute__((ext_vector_type(2)));
typedef float v8f __attribute__((ext_vector_type(8)));

#define NEG_SLOPE 0.2f
#define LN_EPS    1e-5f
#define BB 64
#define KK 256
#define DD 1024
#define HH 2

// ---------------------------------------------------------------------------
// Batched FP32 WMMA GEMM: C[z] = A[z] @ B[z]
// Block tile 128(M) x 64(N), K-step 32, double-buffered LDS filled with
// global_load_async_to_lds_b128/b32 (ASYNCcnt). 8 waves; wave w computes rows
// [w*16, w*16+16) x all 64 cols = 4 16x16 f32 accumulators.
// LDS layouts (stride 36 floats = 144 B: 16B-aligned rows, conflict-free
// 16-row column reads since 36*r mod 64 is a permutation):
//   As[m][k]  (row-major)        -> A frag = ds_load b64 pair
//   Bs[n][k]  (TRANSPOSED)       -> B frag = ds_load_b64, no reg shuffles
// Fragments (ISA 7.12.2, f32):
//   A 16x4 : lane L -> M = L%16, K = 2*(L/16)+v        (2 VGPRs)
//   B 4x16 : lane L -> N = L%16, K = 2*(L/16)+v        (2 VGPRs)
//   C 16x16: lane L, vgpr v -> M = v + 8*(L>=16), N = L%16
// ---------------------------------------------------------------------------
__global__ __launch_bounds__(256)
void gemm_f32_wmma(const float* __restrict__ A, const float* __restrict__ B,
                   float* __restrict__ C, int Kd, int lda, int ldb, int ldc,
                   long sA, long sB, long sC)
{
    __shared__ float As[2][128][36];
    __shared__ float Bs[2][64][36];    // transposed: [n][k]

    const int tid  = threadIdx.x;
    const int lane = tid & 31;
    const int wave = tid >> 5;          // 0..7 -> 16-row strip
    const int l16  = lane & 15;
    const int hi   = lane >> 4;         // 0/1
    const int tileN = blockIdx.x * 64;
    const int tileM = blockIdx.y * 128;

    const float* Ab = A + (long)blockIdx.z * sA;
    const float* Bb = B + (long)blockIdx.z * sB;
    float*       Cb = C + (long)blockIdx.z * sC;

    // --- per-thread async staging descriptors -----------------------------
    // A tile 128x32 = 1024 float4 -> 4 b128 asyncs per thread
    const float* ap[4];
    uint32_t     ao[4];
    #pragma unroll
    for (int t = 0; t < 4; ++t) {
        int idx = tid + t * 256;
        int r = idx >> 3, c4 = idx & 7;              // 8 float4 per row
        ap[t] = Ab + (long)(tileM + r) * lda + c4 * 4;
        ao[t] = (uint32_t)(uintptr_t)&As[0][r][c4 * 4];
    }
    // B tile 32(k)x64(n), stored transposed: 8 b32 asyncs per thread.
    // thread -> n = tid&63, k = (tid>>6) + 4*t ; wave covers 32 consecutive n
    // at fixed k => fully coalesced 128B global requests.
    const int nIdx  = tid & 63;
    const int kbase = tid >> 6;                      // 0..3
    const float* bp = Bb + (long)kbase * ldb + tileN + nIdx;
    const uint32_t bo = (uint32_t)(uintptr_t)&Bs[0][nIdx][kbase];

    constexpr uint32_t ABUF = 128 * 36 * 4;          // byte stride buf0 -> buf1
    constexpr uint32_t BBUF = 64 * 36 * 4;

    // issue one k-block of async global->LDS copies into buffer {a_off,b_off}
    auto issue = [&](uint32_t a_off, uint32_t b_off) {
        #pragma unroll
        for (int t = 0; t < 4; ++t) {
            asm volatile("global_load_async_to_lds_b128 %0, %1, off"
                         :: "v"(ao[t] + a_off), "v"(ap[t]) : "memory");
            ap[t] += 32;
        }
        const float* gp = bp;
        #pragma unroll
        for (int t = 0; t < 8; ++t) {                // k = kbase + 4*t
            asm volatile("global_load_async_to_lds_b32 %0, %1, off"
                         :: "v"(bo + b_off + t * 16), "v"(gp) : "memory");
            gp += (long)ldb * 4;
        }
        bp += (long)ldb * 32;
    };

    v8f c0 = {}, c1 = {}, c2 = {}, c3 = {};

    auto compute = [&](const float (*Asb)[36], const float (*Bsb)[36]) {
        #pragma unroll
        for (int kk = 0; kk < 8; ++kk) {
            const int kb = kk * 4 + hi * 2;
            v2f a, b;
            a.x = Asb[wave * 16 + l16][kb];
            a.y = Asb[wave * 16 + l16][kb + 1];
            b.x = Bsb[l16][kb];      b.y = Bsb[l16][kb + 1];
            c0 = __builtin_amdgcn_wmma_f32_16x16x4_f32(false, a, false, b, (short)0, c0, false, false);
            b.x = Bsb[16 + l16][kb]; b.y = Bsb[16 + l16][kb + 1];
            c1 = __builtin_amdgcn_wmma_f32_16x16x4_f32(false, a, false, b, (short)0, c1, false, false);
            b.x = Bsb[32 + l16][kb]; b.y = Bsb[32 + l16][kb + 1];
            c2 = __builtin_amdgcn_wmma_f32_16x16x4_f32(false, a, false, b, (short)0, c2, false, false);
            b.x = Bsb[48 + l16][kb]; b.y = Bsb[48 + l16][kb + 1];
            c3 = __builtin_amdgcn_wmma_f32_16x16x4_f32(false, a, false, b, (short)0, c3, false, false);
        }
    };

    // prologue: k-block 0 -> buf0
    issue(0, 0);
    asm volatile("s_wait_asynccnt 0x0" ::: "memory");
    __syncthreads();

    const int nblk = Kd >> 5;                        // even for all our calls
    for (int blk = 0; blk < nblk; blk += 2) {
        issue(ABUF, BBUF);                           // prefetch blk+1 -> buf1
        compute(As[0], Bs[0]);
        asm volatile("s_wait_asynccnt 0x0" ::: "memory");
        __syncthreads();
        if (blk + 2 < nblk) issue(0, 0);             // prefetch blk+2 -> buf0
        compute(As[1], Bs[1]);
        asm volatile("s_wait_asynccnt 0x0" ::: "memory");
        __syncthreads();
    }

    const int rowBase = tileM + wave * 16 + hi * 8;
    const int col0    = tileN + l16;
    #pragma unroll
    for (int v = 0; v < 8; ++v) {
        long r = (long)(rowBase + v) * ldc;
        Cb[r + col0]      = c0[v];
        Cb[r + col0 + 16] = c1[v];
        Cb[r + col0 + 32] = c2[v];
        Cb[r + col0 + 48] = c3[v];
    }
}

// ---------------------------------------------------------------------------
// f_i[b,h,k] = dot(h[b,k,:], a[h,:D]); f_j[b,h,k] = dot(h[b,k,:], a[h,D:])
// One block per (b,k); one float4 per thread (D/4 == 256).
// ---------------------------------------------------------------------------
__global__ __launch_bounds__(256)
void attn_feats(const float* __restrict__ h, const float* __restrict__ a,
                float* __restrict__ fi, float* __restrict__ fj)
{
    __shared__ float red[4][256];
    const int row = blockIdx.x;           // b*K + k
    const int tid = threadIdx.x;
    const float4* h4 = (const float4*)(h + (long)row * DD);
    const float4* a4 = (const float4*)a;

    float4 hv  = h4[tid];
    float4 s0v = a4[tid];                  // head0 src
    float4 d0v = a4[(DD >> 2) + tid];      // head0 dst
    float4 s1v = a4[(2 * DD >> 2) + tid];  // head1 src
    float4 d1v = a4[(3 * DD >> 2) + tid];  // head1 dst

    red[0][tid] = hv.x * s0v.x + hv.y * s0v.y + hv.z * s0v.z + hv.w * s0v.w;
    red[1][tid] = hv.x * d0v.x + hv.y * d0v.y + hv.z * d0v.z + hv.w * d0v.w;
    red[2][tid] = hv.x * s1v.x + hv.y * s1v.y + hv.z * s1v.z + hv.w * s1v.w;
    red[3][tid] = hv.x * d1v.x + hv.y * d1v.y + hv.z * d1v.z + hv.w * d1v.w;
    __syncthreads();
    for (int off = 128; off > 0; off >>= 1) {
        if (tid < off) {
            red[0][tid] += red[0][tid + off];
            red[1][tid] += red[1][tid + off];
            red[2][tid] += red[2][tid + off];
            red[3][tid] += red[3][tid + off];
        }
        __syncthreads();
    }
    if (tid == 0) {
        int b = row / KK, k = row % KK;
        fi[(b * HH + 0) * KK + k] = red[0][0];
        fj[(b * HH + 0) * KK + k] = red[1][0];
        fi[(b * HH + 1) * KK + k] = red[2][0];
        fj[(b * HH + 1) * KK + k] = red[3][0];
    }
}

// ---------------------------------------------------------------------------
// Per (b,i): e[h,j] = leaky(f_i[h,i]+f_j[h,j]); mask by adj; softmax over j
// per head (fully-masked row -> 0, matching NaN->0); average the two heads.
// ---------------------------------------------------------------------------
__global__ __launch_bounds__(256)
void attn_softmax(const float* __restrict__ fi, const float* __restrict__ fj,
                  const int* __restrict__ adj, float* __restrict__ attn_avg)
{
    __shared__ float r0[256], r1[256];
    const int row = blockIdx.x;           // b*K + i
    const int b = row / KK, i = row % KK;
    const int j = threadIdx.x;

    const float fi0 = fi[(b * HH + 0) * KK + i];
    const float fi1 = fi[(b * HH + 1) * KK + i];
    const float fj0 = fj[(b * HH + 0) * KK + j];
    const float fj1 = fj[(b * HH + 1) * KK + j];
    const bool m = adj[i * KK + j] != 0;

    const float NINF = -__builtin_inff();
    float e0 = fi0 + fj0; e0 = (e0 >= 0.f) ? e0 : NEG_SLOPE * e0;
    float e1 = fi1 + fj1; e1 = (e1 >= 0.f) ? e1 : NEG_SLOPE * e1;
    if (!m) { e0 = NINF; e1 = NINF; }

    r0[j] = e0; r1[j] = e1; __syncthreads();
    for (int off = 128; off > 0; off >>= 1) {
        if (j < off) { r0[j] = fmaxf(r0[j], r0[j + off]);
                       r1[j] = fmaxf(r1[j], r1[j + off]); }
        __syncthreads();
    }
    const float m0 = r0[0], m1 = r1[0];
    __syncthreads();

    float p0 = (m0 == NINF) ? 0.f : __expf(e0 - m0);
    float p1 = (m1 == NINF) ? 0.f : __expf(e1 - m1);

    r0[j] = p0; r1[j] = p1; __syncthreads();
    for (int off = 128; off > 0; off >>= 1) {
        if (j < off) { r0[j] += r0[j + off]; r1[j] += r1[j + off]; }
        __syncthreads();
    }
    const float s0 = r0[0], s1 = r1[0];
    const float a0 = (s0 > 0.f) ? p0 / s0 : 0.f;
    const float a1 = (s1 > 0.f) ? p1 / s1 : 0.f;
    attn_avg[(long)row * KK + j] = 0.5f * (a0 + a1);
}

// ---------------------------------------------------------------------------
// In-place: out = LayerNorm(out + x) * gamma + beta; one float4 per thread.
// ---------------------------------------------------------------------------
__global__ __launch_bounds__(256)
void residual_ln(float* __restrict__ out, const float* __restrict__ x,
                 const float* __restrict__ gamma, const float* __restrict__ beta)
{
    __shared__ float red[256];
    const int row = blockIdx.x;           // b*K + k
    const int tid = threadIdx.x;
    float4* o4 = (float4*)(out + (long)row * DD);
    const float4* x4 = (const float4*)(x + (long)row * DD);

    float4 xv = x4[tid];
    float4 y  = o4[tid];
    y.x += xv.x; y.y += xv.y; y.z += xv.z; y.w += xv.w;

    red[tid] = y.x + y.y + y.z + y.w; __syncthreads();
    for (int off = 128; off > 0; off >>= 1) {
        if (tid < off) red[tid] += red[tid + off];
        __syncthreads();
    }
    const float mu = red[0] * (1.0f / DD);
    __syncthreads();

    float4 t = make_float4(y.x - mu, y.y - mu, y.z - mu, y.w - mu);
    red[tid] = t.x * t.x + t.y * t.y + t.z * t.z + t.w * t.w; __syncthreads();
    for (int off = 128; off > 0; off >>= 1) {
        if (tid < off) red[tid] += red[tid + off];
        __syncthreads();
    }
    const float rstd = rsqrtf(red[0] * (1.0f / DD) + LN_EPS);

    float4 g = ((const float4*)gamma)[tid];
    float4 be = ((const float4*)beta)[tid];
    o4[tid] = make_float4(t.x * rstd * g.x + be.x, t.y * rstd * g.y + be.y,
                          t.z * rstd * g.z + be.z, t.w * rstd * g.w + be.w);
}

// ---------------------------------------------------------------------------
extern "C" void kernel_launch(void* const* d_in, const int* in_sizes, int n_in,
                              void* d_out, int out_size, void* d_ws, size_t ws_size,
                              hipStream_t stream)
{
    const float* x     = (const float*)d_in[0];
    const int*   adj   = (const int*)  d_in[1];
    const float* W     = (const float*)d_in[2];
    const float* a     = (const float*)d_in[3];
    const float* gamma = (const float*)d_in[4];
    const float* beta  = (const float*)d_in[5];
    float* out = (float*)d_out;

    char* ws = (char*)d_ws;
    float* h    = (float*)ws;                                        // B*K*D (64 MB)
    float* attn = (float*)(ws + (size_t)BB * KK * DD * 4);           // B*K*K (16 MB)
    float* fi   = (float*)(ws + (size_t)BB * KK * DD * 4
                              + (size_t)BB * KK * KK * 4);           // B*H*K
    float* fj   = fi + BB * HH * KK;                                 // B*H*K

    // 1) h = X @ W : M = B*K = 16384, N = D, Kd = D
    gemm_f32_wmma<<<dim3(DD / 64, (BB * KK) / 128, 1), 256, 0, stream>>>(
        x, W, h, /*Kd=*/DD, /*lda=*/DD, /*ldb=*/DD, /*ldc=*/DD, 0L, 0L, 0L);

    // 2) attention features
    attn_feats<<<BB * KK, 256, 0, stream>>>(h, a, fi, fj);

    // 3) masked leaky softmax, head-averaged attention
    attn_softmax<<<BB * KK, 256, 0, stream>>>(fi, fj, adj, attn);

    // 4) out_pre = attn_avg @ h (batched over B): M = K, N = D, Kd = K
    gemm_f32_wmma<<<dim3(DD / 64, KK / 128, BB), 256, 0, stream>>>(
        attn, h, out, /*Kd=*/KK, /*lda=*/KK, /*ldb=*/DD, /*ldc=*/DD,
        (long)KK * KK, (long)KK * DD, (long)KK * DD);

    // 5) residual + LayerNorm (in place on d_out)
    residual_ln<<<BB * KK, 256, 0, stream>>>(out, x, gamma, beta);
}